// ForwardKinematics_24189255811652
// MI455X (gfx1250) — compile-verified
//
#include <hip/hip_runtime.h>
#include <stdint.h>

// ---------------------------------------------------------------------------
// Forward kinematics, SMPL 24-joint tree, N samples.
//  pos_j = R_parent @ offset_j + pos_parent ;  R_j = R_parent @ A_j ; R_0 = I.
// Memory-bound (~302 MB traffic, ~1.4 FLOP/B) -> optimize the data path:
// coalesced async global->LDS staging (CDNA5 GLOBAL_LOAD_ASYNC_TO_LDS_B128,
// non-temporal), per-thread row reads from LDS, results staged back via
// GLOBAL_STORE_ASYNC_FROM_LDS_B128 (non-temporal).
// ---------------------------------------------------------------------------

#define TILE        128      // samples per block == threads per block
#define ROW_F4      54       // 216 floats per sample = 54 float4
#define ROW_F4_PAD  55       // +1 float4 pad -> conflict-free LDS reads
#define OUT_F4      18       // 72 floats per sample = 18 float4
#define OUT_F4_PAD  19

#define N_JOINTS 24

__global__ __launch_bounds__(TILE)
void fk_kernel(const float* __restrict__ ja,     // (N, 216)
               const float* __restrict__ offs,   // (24, 3)
               float* __restrict__ out)          // (N, 24, 3)
{
    __shared__ float4 s_tile[TILE * ROW_F4_PAD];   // 110.0 KB
    __shared__ float4 s_out [TILE * OUT_F4_PAD];   //  38.0 KB

    const unsigned tid = threadIdx.x;
    const unsigned blk = blockIdx.x;

    // Workgroup-relative LDS byte addresses (low 32 bits of the flat address
    // of an LDS object are the DS offset).
    const unsigned ldsTile = (unsigned)(uintptr_t)(&s_tile[0]);
    const unsigned ldsOut  = (unsigned)(uintptr_t)(&s_out[0]);

    // ---- Phase 1: async, fully coalesced copy of the input tile into LDS ---
    // Linear float4 index e over the tile; sample s = e/54 ; LDS gets one
    // float4 of pad per row, so lds_off = (e + s) * 16. Stream is read-once:
    // mark non-temporal to avoid thrashing L2 (226 MB stream vs 192 MB L2).
    {
        const float* gin = ja + (size_t)blk * (TILE * 216);
        #pragma unroll
        for (unsigned i = 0; i < ROW_F4; ++i) {
            unsigned e    = i * TILE + tid;        // consecutive across lanes
            unsigned s    = e / ROW_F4;
            unsigned dst  = ldsTile + (e + s) * 16;
            unsigned voff = e * 16;
            asm volatile("global_load_async_to_lds_b128 %0, %1, %2 th:TH_LOAD_NT"
                         :: "v"(dst), "v"(voff), "s"(gin) : "memory");
        }
    }
    asm volatile("s_wait_asynccnt 0" ::: "memory");
    __syncthreads();

    // ---- Phase 2: per-sample kinematic chain (fully unrolled tree) ---------
    constexpr int PAR[N_JOINTS] =
        {-1, 0, 0, 0, 1, 2, 3, 4, 5, 6, 7, 8, 9, 9, 9, 12, 13, 14, 16, 17, 18, 19, 20, 21};
    // Joints whose rotation is consumed by a child (leaves 10,11,15,22,23 are not).
    constexpr bool NEEDR[N_JOINTS] =
        { true, true, true, true, true, true, true, true, true, true,
          false, false, true, true, true, false, true, true, true, true,
          true, true, false, false };

    const float* row = reinterpret_cast<const float*>(&s_tile[0]) + tid * (ROW_F4_PAD * 4);

    float R[N_JOINTS][9];
    float P[N_JOINTS][3];
    P[0][0] = 0.f; P[0][1] = 0.f; P[0][2] = 0.f;

    #pragma unroll
    for (int j = 1; j < N_JOINTS; ++j) {
        const int p = PAR[j];
        const float ox = offs[j * 3 + 0];
        const float oy = offs[j * 3 + 1];
        const float oz = offs[j * 3 + 2];
        if (p == 0) {
            // Parent rotation is identity: pos = offset, R = A_j.
            P[j][0] = ox; P[j][1] = oy; P[j][2] = oz;
            if (NEEDR[j]) {
                #pragma unroll
                for (int k = 0; k < 9; ++k) R[j][k] = row[j * 9 + k];
            }
        } else {
            #pragma unroll
            for (int r = 0; r < 3; ++r) {
                P[j][r] = fmaf(R[p][r * 3 + 0], ox,
                          fmaf(R[p][r * 3 + 1], oy,
                          fmaf(R[p][r * 3 + 2], oz, P[p][r])));
            }
            if (NEEDR[j]) {
                float a[9];
                #pragma unroll
                for (int k = 0; k < 9; ++k) a[k] = row[j * 9 + k];
                #pragma unroll
                for (int r = 0; r < 3; ++r) {
                    #pragma unroll
                    for (int c = 0; c < 3; ++c) {
                        R[j][r * 3 + c] =
                            fmaf(R[p][r * 3 + 0], a[0 * 3 + c],
                            fmaf(R[p][r * 3 + 1], a[1 * 3 + c],
                                 R[p][r * 3 + 2] * a[2 * 3 + c]));
                    }
                }
            }
        }
    }

    // ---- Phase 3: stage results in LDS (b128 stores), async NT store out ---
    {
        float tmp[OUT_F4 * 4];                       // 72 floats, SROA -> regs
        tmp[0] = 0.f; tmp[1] = 0.f; tmp[2] = 0.f;    // root position
        #pragma unroll
        for (int j = 1; j < N_JOINTS; ++j) {
            tmp[j * 3 + 0] = P[j][0];
            tmp[j * 3 + 1] = P[j][1];
            tmp[j * 3 + 2] = P[j][2];
        }
        float4* orow4 = &s_out[tid * OUT_F4_PAD];
        #pragma unroll
        for (int q = 0; q < OUT_F4; ++q) {
            orow4[q] = make_float4(tmp[q * 4 + 0], tmp[q * 4 + 1],
                                   tmp[q * 4 + 2], tmp[q * 4 + 3]);
        }
    }
    __syncthreads();   // DS stores complete & visible before async-store reads

    {
        float* gout = out + (size_t)blk * (TILE * 72);
        #pragma unroll
        for (unsigned i = 0; i < OUT_F4; ++i) {
            unsigned e    = i * TILE + tid;        // consecutive across lanes
            unsigned s    = e / OUT_F4;
            unsigned src  = ldsOut + (e + s) * 16;
            unsigned voff = e * 16;
            asm volatile("global_store_async_from_lds_b128 %0, %1, %2 th:TH_STORE_NT"
                         :: "v"(voff), "v"(src), "s"(gout) : "memory");
        }
    }
    asm volatile("s_wait_asynccnt 0" ::: "memory");
}

extern "C" void kernel_launch(void* const* d_in, const int* in_sizes, int n_in,
                              void* d_out, int out_size, void* d_ws, size_t ws_size,
                              hipStream_t stream) {
    (void)n_in; (void)out_size; (void)d_ws; (void)ws_size;
    const float* ja   = (const float*)d_in[0];   // (N, 24*9) f32
    const float* offs = (const float*)d_in[1];   // (24, 3)   f32
    float* out        = (float*)d_out;           // (N, 24, 3) f32

    const int N    = in_sizes[0] / (N_JOINTS * 9);   // 262144
    const int nblk = N / TILE;                       // N is a multiple of 128
    fk_kernel<<<nblk, TILE, 0, stream>>>(ja, offs, out);
}